// GraphAutoencoder_31748398252155
// MI455X (gfx1250) — compile-verified
//
#include <hip/hip_runtime.h>
#include <hip/hip_bf16.h>

typedef __attribute__((ext_vector_type(2))) float v2f;
typedef __attribute__((ext_vector_type(8))) float v8f;

#define NNODES 50000
#define NEDGES 800000

// ---------------------------------------------------------------- utilities
__global__ void fill_kernel(float* __restrict__ p, float v, int n) {
    int i = blockIdx.x * blockDim.x + threadIdx.x;
    if (i < n) p[i] = v;
}

__global__ void deg_kernel(const long long* __restrict__ dst,
                           float* __restrict__ deg, int E) {
    int e = blockIdx.x * blockDim.x + threadIdx.x;
    if (e < E) atomicAdd(&deg[dst[e]], 1.0f);
}

__global__ void dis_kernel(float* __restrict__ deg, int n) {
    int i = blockIdx.x * blockDim.x + threadIdx.x;
    if (i < n) deg[i] = rsqrtf(fmaxf(deg[i], 1.0f));
}

// ---------------------------------------------------------------- WMMA GEMM
// H[M,N] = X[M,K] * W[K,N]; fp32 matrix core, one wave per 16x16 C tile.
// A 16x4 layout : lane&15 = M row, (lane>>4)*2 = K pair in {a.x,a.y}
// B 4x16 layout : lane&15 = N col, (lane>>4)*2 = K pair in {b.x,b.y}
// C/D           : VGPR r -> M = r + 8*(lane>>4), N = lane&15
__global__ void gemm_wmma_kernel(const float* __restrict__ X,
                                 const float* __restrict__ W,
                                 float* __restrict__ H,
                                 int M, int K, int N) {
    const int lane = threadIdx.x & 31;
    const int m0 = blockIdx.x << 4;
    const int n0 = blockIdx.y << 4;
    const int mr = lane & 15;
    const int kh = (lane >> 4) << 1;    // 0 or 2

    const float* __restrict__ xrow = X + (size_t)(m0 + mr) * K + kh;
    const float* __restrict__ wcol = W + n0 + mr;

    v8f acc = {};
    for (int k = 0; k < K; k += 4) {
        v2f a;
        a.x = xrow[k];
        a.y = xrow[k + 1];
        v2f b;
        b.x = wcol[(size_t)(k + kh) * N];
        b.y = wcol[(size_t)(k + kh + 1) * N];
        acc = __builtin_amdgcn_wmma_f32_16x16x4_f32(
            /*neg_a=*/false, a, /*neg_b=*/false, b,
            /*c_mod=*/(short)0, acc, /*reuse_a=*/false, /*reuse_b=*/false);
    }

    const int mbase = m0 + ((lane >> 4) << 3);
    float* __restrict__ hp = H + (size_t)mbase * N + n0 + mr;
#pragma unroll
    for (int r = 0; r < 8; ++r)
        hp[(size_t)r * N] = acc[r];
}

// ------------------------------------------------------- aggregation stages
// agg[i,:] = h[i,:] * dis[i]^2   (self-loop contribution; also zero-init)
__global__ void selfloop_kernel(const float* __restrict__ H,
                                const float* __restrict__ dis,
                                float* __restrict__ agg,
                                int nquads, int Dq) {
    int t = blockIdx.x * blockDim.x + threadIdx.x;
    if (t >= nquads) return;
    int node = t / Dq;
    float w = dis[node];
    w = w * w;
    float4 h = ((const float4*)H)[t];
    float4 o;
    o.x = h.x * w; o.y = h.y * w; o.z = h.z * w; o.w = h.w * w;
    ((float4*)agg)[t] = o;
}

// agg[dst,:] += h[src,:] * dis[src]*dis[dst]   (64 or fewer lanes per edge,
// consecutive lanes cover consecutive float4 feature quads -> coalesced)
__global__ void scatter_kernel(const float* __restrict__ H,
                               const long long* __restrict__ src,
                               const long long* __restrict__ dst,
                               const float* __restrict__ dis,
                               float* __restrict__ agg,
                               int E, int D) {
    long long t = (long long)blockIdx.x * blockDim.x + threadIdx.x;
    const int Dq = D >> 2;
    long long e = t / Dq;
    if (e >= E) return;
    int f = (int)(t - e * Dq) << 2;
    long long s = src[e];
    long long d = dst[e];
    float nrm = dis[s] * dis[d];
    float4 h = *(const float4*)(H + (size_t)s * D + f);
    float* o = agg + (size_t)d * D + f;
    atomicAdd(o + 0, h.x * nrm);
    atomicAdd(o + 1, h.y * nrm);
    atomicAdd(o + 2, h.z * nrm);
    atomicAdd(o + 3, h.w * nrm);
}

// out = (relu?)(agg + bias)
__global__ void finalize_kernel(const float* __restrict__ agg,
                                const float* __restrict__ bias,
                                float* __restrict__ out,
                                int nquads, int Dq, int do_relu) {
    int t = blockIdx.x * blockDim.x + threadIdx.x;
    if (t >= nquads) return;
    int f = (t % Dq) << 2;
    float4 v = ((const float4*)agg)[t];
    v.x += bias[f + 0];
    v.y += bias[f + 1];
    v.z += bias[f + 2];
    v.w += bias[f + 3];
    if (do_relu) {
        v.x = fmaxf(v.x, 0.0f);
        v.y = fmaxf(v.y, 0.0f);
        v.z = fmaxf(v.z, 0.0f);
        v.w = fmaxf(v.w, 0.0f);
    }
    ((float4*)out)[t] = v;
}

// ---------------------------------------------------------------- launcher
extern "C" void kernel_launch(void* const* d_in, const int* in_sizes, int n_in,
                              void* d_out, int out_size, void* d_ws, size_t ws_size,
                              hipStream_t stream) {
    const float* x  = (const float*)d_in[0];
    const long long* ei = (const long long*)d_in[1];   // [2, E] int64
    const long long* src = ei;
    const long long* dst = ei + NEDGES;
    const float* W[4] = { (const float*)d_in[2], (const float*)d_in[4],
                          (const float*)d_in[6], (const float*)d_in[8] };
    const float* B[4] = { (const float*)d_in[3], (const float*)d_in[5],
                          (const float*)d_in[7], (const float*)d_in[9] };
    const int Kd[4] = { 128, 256,  64, 256 };   // per-layer in-dim
    const int Nd[4] = { 256,  64, 256, 128 };   // per-layer out-dim

    // workspace layout (256B-aligned slabs)
    char* ws = (char*)d_ws;
    const size_t slab = (size_t)NNODES * 256 * sizeof(float);   // 51.2 MB
    float* dis = (float*)ws;                                    // N floats
    float* P   = (float*)(ws + 256 * ((NNODES * sizeof(float) + 255) / 256));
    float* A   = (float*)((char*)P + slab);
    float* Q   = (float*)((char*)A + slab);

    const int TB = 256;

    // degree (incl. self loop) -> dis = rsqrt(max(deg,1))
    fill_kernel<<<(NNODES + TB - 1) / TB, TB, 0, stream>>>(dis, 1.0f, NNODES);
    deg_kernel<<<(NEDGES + TB - 1) / TB, TB, 0, stream>>>(dst, dis, NEDGES);
    dis_kernel<<<(NNODES + TB - 1) / TB, TB, 0, stream>>>(dis, NNODES);

    const float* layer_in = x;
    for (int l = 0; l < 4; ++l) {
        const int K = Kd[l], N = Nd[l];
        const int Dq = N >> 2;
        const int nquads = NNODES * Dq;
        float* layer_out = (l == 3) ? (float*)d_out : Q;

        // H = X * W   (matrix core)
        dim3 ggrid(NNODES / 16, N / 16);
        gemm_wmma_kernel<<<ggrid, 32, 0, stream>>>(layer_in, W[l], P,
                                                   NNODES, K, N);
        // agg = self-loop term
        selfloop_kernel<<<(nquads + TB - 1) / TB, TB, 0, stream>>>(P, dis, A,
                                                                   nquads, Dq);
        // agg += edge messages
        long long sthreads = (long long)NEDGES * Dq;
        int sblocks = (int)((sthreads + TB - 1) / TB);
        scatter_kernel<<<sblocks, TB, 0, stream>>>(P, src, dst, dis, A,
                                                   NEDGES, N);
        // out = (relu)(agg + b)
        finalize_kernel<<<(nquads + TB - 1) / TB, TB, 0, stream>>>(
            A, B[l], layer_out, nquads, Dq, (l < 3) ? 1 : 0);

        layer_in = layer_out;
    }
}